// DifferentiablePointCloudRenderer_6828998001445
// MI455X (gfx1250) — compile-verified
//
#include <hip/hip_runtime.h>

// ---------------------------------------------------------------------------
// Differentiable point-cloud renderer, MI455X (gfx1250, wave32).
//
// Roofline: ~24 MB point reads + ~77 MB output writes + ~8M f32 atomics that
// stay resident in the 192 MB L2  ->  memory/atomic bound, ~5 us at 23.3 TB/s.
//
// Pipeline (all on `stream`):
//   1) k_zero_ch0 : zero channel-0 of d_out (float4 stores)
//   2) k_minmax   : per-batch z min/max (one 256-thread block per batch, LDS tree)
//   3) k_splat    : WMMA f32_16x16x4 rotation (R(16x4) x P^T(4x16)) +
//                   bilinear scatter via hardware global_atomic_add_f32
//   4) k_bcast    : broadcast channel-0 -> channels 1,2 (float4)
// ---------------------------------------------------------------------------

#define BB 128
#define NP 16384
#define W_ 224
#define HW (224 * 224)      // 50176
#define SPLITS 8            // blocks per batch in splat pass

typedef float v2f __attribute__((ext_vector_type(2)));
typedef float v8f __attribute__((ext_vector_type(8)));

// R = R_el @ R_az :
//   row0 = (  ca,     0,    sa    )
//   row1 = (  sa*se,  ce,  -ca*se )
//   row2 = ( -sa*ce,  se,   ca*ce )
__device__ __forceinline__ void rot_rows(float a, float e,
                                         float r0[3], float r1[3], float r2[3]) {
    float sa = sinf(a), ca = cosf(a), se = sinf(e), ce = cosf(e);
    r0[0] = ca;       r0[1] = 0.0f; r0[2] = sa;
    r1[0] = sa * se;  r1[1] = ce;   r1[2] = -ca * se;
    r2[0] = -sa * ce; r2[1] = se;   r2[2] = ca * ce;
}

// ---- 1) zero channel 0 of output (B,3,H,W) --------------------------------
__global__ void k_zero_ch0(float* __restrict__ out) {
    int i   = blockIdx.x * blockDim.x + threadIdx.x;   // float4 index over B*HW/4
    int per = HW / 4;                                  // 12544
    int b   = i / per;
    int r   = i - b * per;
    ((float4*)out)[(size_t)b * 3 * per + r] = make_float4(0.f, 0.f, 0.f, 0.f);
}

// ---- 2) per-batch z min/max ------------------------------------------------
__global__ void k_minmax(const float* __restrict__ pts,
                         const float* __restrict__ az,
                         const float* __restrict__ el,
                         float* __restrict__ ws) {
    int b = blockIdx.x;
    float r0[3], r1[3], r2[3];
    rot_rows(az[b], el[b], r0, r1, r2);

    const float* p = pts + (size_t)b * NP * 3;
    float zmin = 1e30f, zmax = -1e30f;
    for (int n = threadIdx.x; n < NP; n += blockDim.x) {
        float x = p[n * 3 + 0], y = p[n * 3 + 1], z = p[n * 3 + 2];
        float zz = fmaf(x, r2[0], fmaf(y, r2[1], z * r2[2]));
        zmin = fminf(zmin, zz);
        zmax = fmaxf(zmax, zz);
    }
    __shared__ float smin[256];
    __shared__ float smax[256];
    smin[threadIdx.x] = zmin;
    smax[threadIdx.x] = zmax;
    __syncthreads();
    for (int s = 128; s > 0; s >>= 1) {
        if (threadIdx.x < (unsigned)s) {
            smin[threadIdx.x] = fminf(smin[threadIdx.x], smin[threadIdx.x + s]);
            smax[threadIdx.x] = fmaxf(smax[threadIdx.x], smax[threadIdx.x + s]);
        }
        __syncthreads();
    }
    if (threadIdx.x == 0) {
        ws[b]      = smin[0];
        ws[BB + b] = smax[0];
    }
}

// ---- 3) WMMA rotate + bilinear splat --------------------------------------
// A (16x4): rows 0..2 = R, rest 0.  Lane L<16: a0=A[L][0], a1=A[L][1];
//                                   lane L+16: a0=A[L][2], a1=A[L][3]=0.
// B (4x16): B[i][n] = point n coord i.  VGPR0: lanes0-15=K0(x), lanes16-31=K2(z);
//                                       VGPR1: lanes0-15=K1(y), lanes16-31=K3(0).
// D (16x16): VGPR r, lane L<16 = D[r][L]  ->  lane L holds (x,y,z) of point L
//            in d[0],d[1],d[2]; rows 3..15 are zero.
__global__ void k_splat(const float* __restrict__ pts,
                        const float* __restrict__ az,
                        const float* __restrict__ el,
                        const float* __restrict__ ws,
                        float* __restrict__ out) {
    const int b    = blockIdx.x / SPLITS;
    const int seg  = blockIdx.x % SPLITS;
    const int lane = threadIdx.x & 31;
    const int wave = threadIdx.x >> 5;
    const bool hi  = lane >= 16;

    float r0[3], r1[3], r2[3];
    rot_rows(az[b], el[b], r0, r1, r2);

    // Loop-invariant A fragment (rotation matrix).
    v2f afrag;
    {
        int j = lane & 15;
        float a0 = 0.0f, a1 = 0.0f;
        if (j == 0) { a0 = hi ? r0[2] : r0[0]; a1 = hi ? 0.0f : r0[1]; }
        if (j == 1) { a0 = hi ? r1[2] : r1[0]; a1 = hi ? 0.0f : r1[1]; }
        if (j == 2) { a0 = hi ? r2[2] : r2[0]; a1 = hi ? 0.0f : r2[1]; }
        afrag[0] = a0; afrag[1] = a1;
    }

    const float zmin  = ws[b];
    const float denom = ws[BB + b] - zmin + 1e-6f;
    const float scale = 0.7f / denom;           // hoisted: one divide per thread

    const float* p  = pts + (size_t)b * NP * 3;
    float*       ob = out + (size_t)b * 3 * HW;

    const int chunkBase = seg * (NP / SPLITS) + wave * (NP / SPLITS / 8); // 256 pts/wave
    const int loadOff   = hi ? 1 : 0;           // lo lanes read (x,y); hi read (y,z)
    v8f czero = {};

    for (int it = 0; it < 8; ++it) {            // 8 iters x 32 pts = 256 pts
        const int ptBase = chunkBase + it * 32;
        const int pl     = ptBase + (lane & 15);

        // Uniform-shape loads for every lane (always in-bounds), select after:
        //   lo lane: (e0,e1) = (x,y) -> b = (x, y)
        //   hi lane: (e0,e1) = (y,z) -> b = (z, 0)
        const float* sA = p + pl * 3 + loadOff;
        float eA0 = sA[0], eA1 = sA[1];
        const float* sB = p + (pl + 16) * 3 + loadOff;
        float eB0 = sB[0], eB1 = sB[1];

        v2f bA, bB;
        bA[0] = hi ? eA1 : eA0;
        bA[1] = hi ? 0.0f : eA1;
        bB[0] = hi ? eB1 : eB0;
        bB[1] = hi ? 0.0f : eB1;

        v8f dA = __builtin_amdgcn_wmma_f32_16x16x4_f32(false, afrag, false, bA,
                                                       (short)0, czero, false, false);
        v8f dB = __builtin_amdgcn_wmma_f32_16x16x4_f32(false, afrag, false, bB,
                                                       (short)0, czero, false, false);

        // Move set-B results (lanes 0-15) into the upper half-wave so every
        // lane splats exactly one point: point index = ptBase + lane.
        float xs = __shfl(dB[0], lane & 15, 32);
        float ys = __shfl(dB[1], lane & 15, 32);
        float zs = __shfl(dB[2], lane & 15, 32);
        float x = hi ? xs : dA[0];
        float y = hi ? ys : dA[1];
        float z = hi ? zs : dA[2];

        // Bilinear splat (matches reference: masked taps contribute 0 -> skip).
        float px  = fmaf(x, 112.0f, 111.5f);
        float py  = fmaf(y, 112.0f, 111.5f);
        float px1 = floorf(px), py1 = floorf(py);
        float fx  = px - px1,   fy  = py - py1;
        int   ix  = (int)px1,   iy  = (int)py1;
        if (ix >= 0 && iy >= 0 && (ix + 1) < W_ && (iy + 1) < W_) {
            float feat = fmaf(z - zmin, scale, 0.3f);
            float gx   = 1.0f - fx, gy = 1.0f - fy;
            float* q = ob + iy * W_ + ix;
            unsafeAtomicAdd(q,           gx * gy * feat);  // w11 @ (y1,x1)
            unsafeAtomicAdd(q + W_,      gx * fy * feat);  // w12 @ (y2,x1)
            unsafeAtomicAdd(q + 1,       fx * gy * feat);  // w21 @ (y1,x2)
            unsafeAtomicAdd(q + W_ + 1,  fx * fy * feat);  // w22 @ (y2,x2)
        }
    }
}

// ---- 4) broadcast channel 0 -> channels 1,2 --------------------------------
__global__ void k_bcast(float* __restrict__ out) {
    int i   = blockIdx.x * blockDim.x + threadIdx.x;   // float4 index over B*HW/4
    int per = HW / 4;
    int b   = i / per;
    int r   = i - b * per;
    float4* o = (float4*)out;
    size_t base = (size_t)b * 3 * per;
    float4 v = o[base + r];
    o[base + per + r]     = v;
    o[base + 2 * per + r] = v;
}

extern "C" void kernel_launch(void* const* d_in, const int* in_sizes, int n_in,
                              void* d_out, int out_size, void* d_ws, size_t ws_size,
                              hipStream_t stream) {
    const float* pts = (const float*)d_in[0];   // (128,16384,3) f32
    const float* az  = (const float*)d_in[1];   // (128,) f32
    const float* el  = (const float*)d_in[2];   // (128,) f32
    float* out = (float*)d_out;                 // (128,3,224,224) f32
    float* ws  = (float*)d_ws;                  // [0..127]=zmin, [128..255]=zmax

    const int n4blocks = (BB * HW / 4) / 256;   // 6272

    k_zero_ch0<<<n4blocks, 256, 0, stream>>>(out);
    k_minmax <<<BB, 256, 0, stream>>>(pts, az, el, ws);
    k_splat  <<<BB * SPLITS, 256, 0, stream>>>(pts, az, el, ws, out);
    k_bcast  <<<n4blocks, 256, 0, stream>>>(out);
}